// HaarDWTDecoder_34308198761077
// MI455X (gfx1250) — compile-verified
//
#include <hip/hip_runtime.h>

// Inverse 2-level Haar DWT with zero details == 4x4 nearest-neighbor upsample * 0.25.
// Pure data movement: 25 MB in, 403 MB out -> store-bandwidth bound on MI455X
// (23.3 TB/s => ~18us floor). Optimized for the memory path: register reuse of
// the source, 128-bit coalesced nontemporal stores, gfx1250 global_prefetch.

typedef __attribute__((ext_vector_type(4))) float v4f;

namespace {
constexpr int kBatch   = 128;
constexpr int kC       = 3;
constexpr int kCAH     = 128;
constexpr int kCAW     = 128;
constexpr int kH       = 512;
constexpr int kW       = 512;
constexpr int kPlaneRows = kBatch * kC * kCAH;   // 49152 blocks, one per cA row
}

__global__ __launch_bounds__(128)
void haar_iwt_upsample4(const float* __restrict__ z, float* __restrict__ out) {
  const int pr = blockIdx.x;      // flat (b*3 + c)*128 + ys
  const int t  = threadIdx.x;     // cA column 0..127

  const int ys    = pr & 127;
  const int plane = pr >> 7;      // b*3 + c

  const long long srcBase = (long long)plane * (kCAH * kCAW) + (long long)ys * kCAW;

  // Single coalesced 4B load per thread; source element is never needed again
  // by anyone -> nontemporal (last-use) load.
  float v = 0.25f * __builtin_nontemporal_load(z + srcBase + t);
  v4f q = {v, v, v, v};

  // Pull the next cA row toward the caches while our 4 stores drain
  // (guarded so the final row never prefetches past the 25 MB input buffer).
  if (pr + 1 < kPlaneRows) {
    __builtin_prefetch(z + srcBase + kCAW + t, 0, 1);
  }

  // Each thread writes a float4 (its cA value replicated 4x along W) into the
  // 4 consecutive output rows it feeds (replication along H). 128 lanes x 16B
  // = one contiguous 2KB row segment per store -> perfectly coalesced b128.
  float*   dstPlane = out + (long long)plane * (kH * kW);
  const int y0      = ys << 2;
  v4f*     dst      = reinterpret_cast<v4f*>(dstPlane + (long long)y0 * kW) + t;

#pragma unroll
  for (int r = 0; r < 4; ++r) {
    // 403 MB write-once stream > 192 MB L2 -> nontemporal store hint.
    __builtin_nontemporal_store(q, dst);
    dst += kW / 4;   // next output row (pitch 512 floats = 128 float4)
  }
}

extern "C" void kernel_launch(void* const* d_in, const int* in_sizes, int n_in,
                              void* d_out, int out_size, void* d_ws, size_t ws_size,
                              hipStream_t stream) {
  const float* z   = (const float*)d_in[0];
  float*       out = (float*)d_out;
  (void)in_sizes; (void)n_in; (void)out_size; (void)d_ws; (void)ws_size;

  haar_iwt_upsample4<<<dim3(kPlaneRows), dim3(128), 0, stream>>>(z, out);
}